// VNDgcnn_30760555774122
// MI455X (gfx1250) — compile-verified
//
#include <hip/hip_runtime.h>
#include <math.h>

#define B_ 4
#define C_ 21
#define N_ 4096
#define K_ 20
#define CD 63            // C_ * 3 flattened feature dim
#define CDP 64           // padded feature dim for WMMA (k=4 x 16 steps)
#define EPSF 1e-6f
#define BN_EPSF 1e-5f
#define NSLOPE 0.2f

typedef float v2f __attribute__((ext_vector_type(2)));
typedef float v8f __attribute__((ext_vector_type(8)));

// ---------------------------------------------------------------- utilities
__global__ void zero_kernel(float* __restrict__ p, int n) {
  int t = blockIdx.x * blockDim.x + threadIdx.x;
  if (t < n) p[t] = 0.0f;
}

// Transpose (B,63,N) -> (B,N,64) with channel 63 zero-padded; fuse xx[b,n].
__global__ void pad_transpose_kernel(const float* __restrict__ xf, float* __restrict__ xfT,
                                     float* __restrict__ xx) {
  int t = blockIdx.x * blockDim.x + threadIdx.x;   // B_*N_ threads exactly
  int b = t / N_, n = t % N_;
  const float* src = xf + (size_t)b * CD * N_ + n;
  float* dst = xfT + (size_t)t * CDP;
  float s = 0.0f;
  for (int c = 0; c < CD; ++c) {
    float v = src[(size_t)c * N_];                  // coalesced across lanes
    dst[c] = v;
    s += v * v;
  }
  dst[CD] = 0.0f;
  xx[t] = s;
}

// ------------------------------------------------------- kNN via WMMA f32 GEMM
// One wave per 32-row tile: two A register sets share one B tile per 16-column
// step; two independent accumulator chains interleave on the XDL pipe.
// B tile preloaded to registers so loads clause with a single wait.
// Top-20 selection fused in LDS, one row per lane (all 32 lanes active).
// pd[m][n] = 2*dot(x_m,x_n) - xx[m] - xx[n].
__global__ __launch_bounds__(32) void knn_wmma_kernel(const float* __restrict__ xfT,
                                                      const float* __restrict__ xx,
                                                      int* __restrict__ idx_out) {
  const int b = blockIdx.y;
  const int row_base = blockIdx.x * 32;
  const int lane = threadIdx.x;
  const float* xftb = xfT + (size_t)b * N_ * CDP;
  const float* xxb  = xx + b * N_;

  __shared__ float s_tile[32][17];
  __shared__ float s_topv[32][20];
  __shared__ int   s_topi[32][20];

  for (int j = 0; j < 20; ++j) { s_topv[lane][j] = -3.0e38f; s_topi[lane][j] = 0; }
  __syncthreads();

  // A layout (16x4 f32): lanes 0-15 -> M=lane, VGPR0/1 = K0/K1;
  //                      lanes16-31 -> M=lane-16, VGPR0/1 = K2/K3.
  const int mrow0 = row_base + (lane & 15);        // rows 0-15 of tile
  const int mrow1 = mrow0 + 16;                    // rows 16-31 of tile
  const int khalf = (lane >> 4) * 2;
  const float* arow0 = xftb + (size_t)mrow0 * CDP + khalf;
  const float* arow1 = xftb + (size_t)mrow1 * CDP + khalf;
  v2f a0[16], a1[16];
  #pragma unroll
  for (int kk = 0; kk < 16; ++kk) {
    a0[kk] = *(const v2f*)(arow0 + kk * 4);        // global_load_b64
    a1[kk] = *(const v2f*)(arow1 + kk * 4);
  }
  const float xxrow = xxb[row_base + lane];        // per-lane selection row norm

  for (int ct = 0; ct < N_ / 16; ++ct) {
    const int col_base = ct * 16;
    const int ncol = col_base + (lane & 15);
    const float* brow = xftb + (size_t)ncol * CDP + khalf;

    v2f b_reg[16];
    #pragma unroll
    for (int kk = 0; kk < 16; ++kk)                // clause of 16 b64 loads
      b_reg[kk] = *(const v2f*)(brow + kk * 4);

    v8f acc0 = {}, acc1 = {};
    #pragma unroll
    for (int kk = 0; kk < 16; ++kk) {              // two independent MMA chains
      acc0 = __builtin_amdgcn_wmma_f32_16x16x4_f32(false, a0[kk], false, b_reg[kk],
                                                   (short)0, acc0, false, false);
      acc1 = __builtin_amdgcn_wmma_f32_16x16x4_f32(false, a1[kk], false, b_reg[kk],
                                                   (short)0, acc1, false, false);
    }

    // C layout: lane holds column N = lane&15, rows M = v + 8*(lane>>4)
    const float xxc = xxb[ncol];
    const int mbase = (lane >> 4) * 8;
    #pragma unroll
    for (int v = 0; v < 8; ++v) {
      s_tile[mbase + v][lane & 15]      = 2.0f * acc0[v] - xxc;
      s_tile[16 + mbase + v][lane & 15] = 2.0f * acc1[v] - xxc;
    }
    __syncthreads();

    // selection: one row per lane, 16 candidates
    for (int nn = 0; nn < 16; ++nn) {
      float val = s_tile[lane][nn] - xxrow;
      if (val > s_topv[lane][19]) {                // strict: ties keep earlier index
        int pos = 19;
        while (pos > 0 && s_topv[lane][pos - 1] < val) {
          s_topv[lane][pos] = s_topv[lane][pos - 1];
          s_topi[lane][pos] = s_topi[lane][pos - 1];
          --pos;
        }
        s_topv[lane][pos] = val;
        s_topi[lane][pos] = col_base + nn;
      }
    }
    __syncthreads();
  }

  int* op = idx_out + ((size_t)b * N_ + row_base + lane) * K_;
  for (int j = 0; j < K_; ++j) op[j] = s_topi[lane][j];
}

// -------------------------------------------- VN edge layer: stats pass
// thread = (b, n, k, o): p[o][:] = sum_c wf[o,c]*(nb-ctr) + wf[o,21+c]*ctr
__global__ void edge_stats_kernel(const float* __restrict__ xsrc, const int* __restrict__ knn,
                                  const float* __restrict__ wf, float* __restrict__ stats) {
  __shared__ float s_acc[128];
  const int tid = threadIdx.x;
  if (tid < 128) s_acc[tid] = 0.0f;
  __syncthreads();

  int t = blockIdx.x * blockDim.x + tid;       // exact multiple of 256
  int o = t % C_;
  int r = t / C_;
  int k = r % K_;  r /= K_;
  int n = r % N_;
  int b = r / N_;
  int nbid = knn[((size_t)b * N_ + n) * K_ + k];
  const float* xb  = xsrc + (size_t)b * CD * N_;
  const float* wfr = wf + o * (2 * C_);
  float p0 = 0.f, p1 = 0.f, p2 = 0.f;
  for (int c = 0; c < C_; ++c) {
    const float* base = xb + (size_t)(c * 3) * N_;
    float c0 = base[n],      c1 = base[N_ + n],      c2 = base[2 * N_ + n];
    float e0 = base[nbid] - c0, e1 = base[N_ + nbid] - c1, e2 = base[2 * N_ + nbid] - c2;
    float wa = wfr[c], wb = wfr[C_ + c];
    p0 += wa * e0 + wb * c0;  p1 += wa * e1 + wb * c1;  p2 += wa * e2 + wb * c2;
  }
  float norm = sqrtf(p0 * p0 + p1 * p1 + p2 * p2) + EPSF;
  atomicAdd(&s_acc[o], norm);
  atomicAdd(&s_acc[64 + o], norm * norm);
  __syncthreads();
  if (tid < 128) atomicAdd(&stats[tid], s_acc[tid]);
}

// -------------------------------------------- VN edge layer: apply + mean over K
__global__ void edge_apply_kernel(const float* __restrict__ xsrc, const int* __restrict__ knn,
                                  const float* __restrict__ wf, const float* __restrict__ wd,
                                  const float* __restrict__ gam, const float* __restrict__ bet,
                                  const float* __restrict__ stats, float* __restrict__ fout) {
  int t = blockIdx.x * blockDim.x + threadIdx.x;
  int o = t % C_;
  int r = t / C_;
  int k = r % K_;  r /= K_;
  int n = r % N_;
  int b = r / N_;
  int nbid = knn[((size_t)b * N_ + n) * K_ + k];
  const float* xb  = xsrc + (size_t)b * CD * N_;
  const float* wfr = wf + o * (2 * C_);
  const float* wdr = wd + o * (2 * C_);
  float p0=0.f,p1=0.f,p2=0.f,d0=0.f,d1=0.f,d2=0.f;
  for (int c = 0; c < C_; ++c) {
    const float* base = xb + (size_t)(c * 3) * N_;
    float c0 = base[n],      c1 = base[N_ + n],      c2 = base[2 * N_ + n];
    float e0 = base[nbid] - c0, e1 = base[N_ + nbid] - c1, e2 = base[2 * N_ + nbid] - c2;
    float wa = wfr[c], wb = wfr[C_ + c];
    p0 += wa * e0 + wb * c0;  p1 += wa * e1 + wb * c1;  p2 += wa * e2 + wb * c2;
    float va = wdr[c], vb = wdr[C_ + c];
    d0 += va * e0 + vb * c0;  d1 += va * e1 + vb * c1;  d2 += va * e2 + vb * c2;
  }
  float norm = sqrtf(p0 * p0 + p1 * p1 + p2 * p2) + EPSF;
  const float cnt = (float)B_ * (float)N_ * (float)K_;
  float mu  = stats[o] / cnt;
  float var = stats[64 + o] / cnt - mu * mu;
  float nbn = (norm - mu) * rsqrtf(var + BN_EPSF) * gam[o] + bet[o];
  float s = nbn / norm;
  float q0 = p0 * s, q1 = p1 * s, q2 = p2 * s;
  float dot = q0 * d0 + q1 * d1 + q2 * d2;
  float dns = d0 * d0 + d1 * d1 + d2 * d2;
  float f = dot / (dns + EPSF);
  float l0, l1, l2;
  if (dot >= 0.0f) { l0 = q0; l1 = q1; l2 = q2; }
  else { l0 = q0 - f * d0; l1 = q1 - f * d1; l2 = q2 - f * d2; }
  float r0 = NSLOPE * q0 + (1.0f - NSLOPE) * l0;
  float r1 = NSLOPE * q1 + (1.0f - NSLOPE) * l1;
  float r2 = NSLOPE * q2 + (1.0f - NSLOPE) * l2;
  const float invK = 1.0f / (float)K_;
  float* fo = fout + (size_t)(b * C_ + o) * 3 * N_ + n;
  atomicAdd(fo,           r0 * invK);
  atomicAdd(fo + N_,      r1 * invK);
  atomicAdd(fo + 2 * N_,  r2 * invK);
}

// -------------------------------------------- final layer (no K): h = [x;f1;f2]
__device__ __forceinline__ const float* hsrc(const float* x, const float* f1, const float* f2,
                                             int c, int& cc) {
  if (c < C_)      { cc = c;          return x;  }
  if (c < 2 * C_)  { cc = c - C_;     return f1; }
  cc = c - 2 * C_;                    return f2;
}

__global__ void final_stats_kernel(const float* __restrict__ x, const float* __restrict__ f1,
                                   const float* __restrict__ f2, const float* __restrict__ wfl,
                                   float* __restrict__ stats) {
  __shared__ float s_acc[128];
  const int tid = threadIdx.x;
  if (tid < 128) s_acc[tid] = 0.0f;
  __syncthreads();
  int t = blockIdx.x * blockDim.x + tid;       // B_*N_*42, exact multiple of 256
  int o = t % (2 * C_);
  int r = t / (2 * C_);
  int n = r % N_;
  int b = r / N_;
  const float* wr = wfl + o * CD;
  float p0 = 0.f, p1 = 0.f, p2 = 0.f;
  for (int c = 0; c < CD; ++c) {
    int cc; const float* src = hsrc(x, f1, f2, c, cc);
    const float* base = src + ((size_t)b * C_ + cc) * 3 * N_ + n;
    float w = wr[c];
    p0 += w * base[0];  p1 += w * base[N_];  p2 += w * base[2 * N_];
  }
  float norm = sqrtf(p0 * p0 + p1 * p1 + p2 * p2) + EPSF;
  atomicAdd(&s_acc[o], norm);
  atomicAdd(&s_acc[64 + o], norm * norm);
  __syncthreads();
  if (tid < 128) atomicAdd(&stats[tid], s_acc[tid]);
}

__global__ void final_apply_kernel(const float* __restrict__ x, const float* __restrict__ f1,
                                   const float* __restrict__ f2, const float* __restrict__ wfl,
                                   const float* __restrict__ wdl, const float* __restrict__ gam,
                                   const float* __restrict__ bet, const float* __restrict__ stats,
                                   float* __restrict__ out) {
  int t = blockIdx.x * blockDim.x + threadIdx.x;
  int o = t % (2 * C_);
  int r = t / (2 * C_);
  int n = r % N_;
  int b = r / N_;
  const float* wfr = wfl + o * CD;
  const float* wdr = wdl + o * CD;
  float p0=0.f,p1=0.f,p2=0.f,d0=0.f,d1=0.f,d2=0.f;
  for (int c = 0; c < CD; ++c) {
    int cc; const float* src = hsrc(x, f1, f2, c, cc);
    const float* base = src + ((size_t)b * C_ + cc) * 3 * N_ + n;
    float h0 = base[0], h1 = base[N_], h2 = base[2 * N_];
    float wa = wfr[c];  p0 += wa * h0;  p1 += wa * h1;  p2 += wa * h2;
    float va = wdr[c];  d0 += va * h0;  d1 += va * h1;  d2 += va * h2;
  }
  float norm = sqrtf(p0 * p0 + p1 * p1 + p2 * p2) + EPSF;
  const float cnt = (float)B_ * (float)N_;
  float mu  = stats[o] / cnt;
  float var = stats[64 + o] / cnt - mu * mu;
  float nbn = (norm - mu) * rsqrtf(var + BN_EPSF) * gam[o] + bet[o];
  float s = nbn / norm;
  float q0 = p0 * s, q1 = p1 * s, q2 = p2 * s;
  float dot = q0 * d0 + q1 * d1 + q2 * d2;
  float dns = d0 * d0 + d1 * d1 + d2 * d2;
  float f = dot / (dns + EPSF);
  float l0, l1, l2;
  if (dot >= 0.0f) { l0 = q0; l1 = q1; l2 = q2; }
  else { l0 = q0 - f * d0; l1 = q1 - f * d1; l2 = q2 - f * d2; }
  float* op = out + (size_t)(b * (2 * C_) + o) * 3 * N_ + n;
  op[0]      = NSLOPE * q0 + (1.0f - NSLOPE) * l0;
  op[N_]     = NSLOPE * q1 + (1.0f - NSLOPE) * l1;
  op[2 * N_] = NSLOPE * q2 + (1.0f - NSLOPE) * l2;
}

// ---------------------------------------------------------------- launcher
extern "C" void kernel_launch(void* const* d_in, const int* in_sizes, int n_in,
                              void* d_out, int out_size, void* d_ws, size_t ws_size,
                              hipStream_t stream) {
  const float* x   = (const float*)d_in[0];
  const float* wf0 = (const float*)d_in[1];
  const float* wd0 = (const float*)d_in[2];
  const float* g0  = (const float*)d_in[3];
  const float* b0  = (const float*)d_in[4];
  const float* wf1 = (const float*)d_in[5];
  const float* wd1 = (const float*)d_in[6];
  const float* g1  = (const float*)d_in[7];
  const float* b1  = (const float*)d_in[8];
  const float* wfl = (const float*)d_in[9];
  const float* wdl = (const float*)d_in[10];
  const float* gl  = (const float*)d_in[11];
  const float* bl  = (const float*)d_in[12];
  float* out = (float*)d_out;

  // workspace layout (floats): xfT | xx | idx | f1 | f2 | stats  (~14 MB)
  float* ws    = (float*)d_ws;
  float* xfT   = ws;                                   // B*N*64
  float* xx    = xfT + (size_t)B_ * N_ * CDP;          // B*N
  int*   idx   = (int*)(xx + B_ * N_);                 // B*N*K
  float* f1    = xx + B_ * N_ + B_ * N_ * K_;          // B*21*3*N
  float* f2    = f1 + (size_t)B_ * C_ * 3 * N_;
  float* stats = f2 + (size_t)B_ * C_ * 3 * N_;        // 128

  const dim3 blk256(256);
  const int featN = B_ * C_ * 3 * N_;                  // 1,032,192
  const int edgeT = B_ * N_ * K_ * C_;                 // 6,881,280 (x256 exact)
  const int finT  = B_ * N_ * 2 * C_;                  // 688,128  (x256 exact)

  // ---- layer 0
  pad_transpose_kernel<<<dim3(B_ * N_ / 256), blk256, 0, stream>>>(x, xfT, xx);
  knn_wmma_kernel<<<dim3(N_ / 32, B_), dim3(32), 0, stream>>>(xfT, xx, idx);
  zero_kernel<<<dim3(1), dim3(128), 0, stream>>>(stats, 128);
  edge_stats_kernel<<<dim3(edgeT / 256), blk256, 0, stream>>>(x, idx, wf0, stats);
  zero_kernel<<<dim3(featN / 256), blk256, 0, stream>>>(f1, featN);
  edge_apply_kernel<<<dim3(edgeT / 256), blk256, 0, stream>>>(x, idx, wf0, wd0, g0, b0, stats, f1);

  // ---- layer 1
  pad_transpose_kernel<<<dim3(B_ * N_ / 256), blk256, 0, stream>>>(f1, xfT, xx);
  knn_wmma_kernel<<<dim3(N_ / 32, B_), dim3(32), 0, stream>>>(xfT, xx, idx);
  zero_kernel<<<dim3(1), dim3(128), 0, stream>>>(stats, 128);
  edge_stats_kernel<<<dim3(edgeT / 256), blk256, 0, stream>>>(f1, idx, wf1, stats);
  zero_kernel<<<dim3(featN / 256), blk256, 0, stream>>>(f2, featN);
  edge_apply_kernel<<<dim3(edgeT / 256), blk256, 0, stream>>>(f1, idx, wf1, wd1, g1, b1, stats, f2);

  // ---- final layer
  zero_kernel<<<dim3(1), dim3(128), 0, stream>>>(stats, 128);
  final_stats_kernel<<<dim3(finT / 256), blk256, 0, stream>>>(x, f1, f2, wfl, stats);
  final_apply_kernel<<<dim3(finT / 256), blk256, 0, stream>>>(x, f1, f2, wfl, wdl, gl, bl, stats, out);
}